// GCNLayer_84043920048503
// MI455X (gfx1250) — compile-verified
//
#include <hip/hip_runtime.h>
#include <stdint.h>

// ---------------------------------------------------------------------------
// CDNA5 async global->LDS helpers (gfx1250).
// Builtin path if the toolchain declares it, inline-asm fallback otherwise.
// ---------------------------------------------------------------------------
#define __AS(n) __attribute__((address_space(n)))

__device__ __forceinline__ void cdna5_async_load_b32(const void* gsrc, void* lds_dst) {
#if __has_builtin(__builtin_amdgcn_global_load_async_to_lds_b32)
    __builtin_amdgcn_global_load_async_to_lds_b32(
        (__AS(1) int*)(uintptr_t)gsrc,
        (__AS(3) int*)(uintptr_t)lds_dst,
        /*imm offset*/ 0, /*cpol*/ 0);
#else
    uint32_t lds_off = (uint32_t)(uintptr_t)lds_dst;   // low 32 bits = LDS byte offset
    asm volatile("global_load_async_to_lds_b32 %0, %1, off"
                 :: "v"(lds_off), "v"(gsrc) : "memory");
#endif
}

__device__ __forceinline__ void cdna5_wait_async0() {
#if __has_builtin(__builtin_amdgcn_s_wait_asynccnt)
    __builtin_amdgcn_s_wait_asynccnt(0);
#else
    asm volatile("s_wait_asynccnt 0" ::: "memory");
#endif
}

// ---------------------------------------------------------------------------
// Kernels
// ---------------------------------------------------------------------------
constexpr int D_FEAT     = 64;     // features per node (fixed by reference)
constexpr int TILE_EDGES = 256;    // edges staged per block
constexpr int BLOCK      = 256;    // 8 wave32s

// Zero d_out (harness poisons once; we accumulate, so zero on every launch).
__global__ __launch_bounds__(256)
void gcn_zero_out(float* __restrict__ out, int n) {
    int i = (blockIdx.x * blockDim.x + threadIdx.x) * 4;
    if (i + 3 < n) {
        *(float4*)(out + i) = make_float4(0.f, 0.f, 0.f, 0.f);
    } else {
        for (; i < n; ++i) out[i] = 0.f;
    }
}

// SpMM scatter: each 16-lane group owns one edge per iteration.
// Lane l loads float4 #l of the gathered embeds row (16 lanes x 16B = one
// fully-coalesced 256B row), scales by val, and atomically adds into out[row].
__global__ __launch_bounds__(BLOCK)
void gcn_spmm_scatter(const int*   __restrict__ rows,
                      const int*   __restrict__ cols,
                      const float* __restrict__ vals,
                      const float* __restrict__ embeds,
                      float*       __restrict__ out,
                      int E) {
    __shared__ int   s_rows[TILE_EDGES];
    __shared__ int   s_cols[TILE_EDGES];
    __shared__ float s_vals[TILE_EDGES];

    const int tid  = threadIdx.x;
    const int base = blockIdx.x * TILE_EDGES;
    const int e0   = base + tid;

    // Stage this block's edge metadata into LDS via async DMA (ASYNCcnt path).
    if (e0 < E) {
        cdna5_async_load_b32(rows + e0, &s_rows[tid]);
        cdna5_async_load_b32(cols + e0, &s_cols[tid]);
        cdna5_async_load_b32(vals + e0, &s_vals[tid]);
    }
    cdna5_wait_async0();
    __syncthreads();

    const int g = tid >> 4;   // half-wave group id: 0..15
    const int l = tid & 15;   // feature chunk id within edge: float4 #l

    const float4* embeds4 = (const float4*)embeds;

    #pragma unroll
    for (int it = 0; it < TILE_EDGES / 16; ++it) {
        const int el = it * 16 + g;
        const int eg = base + el;
        if (eg < E) {
            const int   row = s_rows[el];   // same addr across 16 lanes -> LDS broadcast
            const int   col = s_cols[el];
            const float val = s_vals[el];

            // Coalesced 256B gather of one embeds row (mostly L2-resident: 25.6MB << 192MB L2).
            float4 f = embeds4[(size_t)col * (D_FEAT / 4) + l];

            float* dst = out + (size_t)row * D_FEAT + l * 4;
            atomicAdd(dst + 0, f.x * val);   // no-return global_atomic_add_f32
            atomicAdd(dst + 1, f.y * val);
            atomicAdd(dst + 2, f.z * val);
            atomicAdd(dst + 3, f.w * val);
        }
    }
}

// ---------------------------------------------------------------------------
// Launch
// ---------------------------------------------------------------------------
extern "C" void kernel_launch(void* const* d_in, const int* in_sizes, int n_in,
                              void* d_out, int out_size, void* d_ws, size_t ws_size,
                              hipStream_t stream) {
    const int*   rows   = (const int*)  d_in[0];
    const int*   cols   = (const int*)  d_in[1];
    const float* vals   = (const float*)d_in[2];
    const float* embeds = (const float*)d_in[3];
    float*       out    = (float*)d_out;

    const int E = in_sizes[0];

    // 1) zero the accumulator
    const int zthreads = 256;
    const int zblocks  = (out_size + zthreads * 4 - 1) / (zthreads * 4);
    gcn_zero_out<<<zblocks, zthreads, 0, stream>>>(out, out_size);

    // 2) scatter SpMM
    const int blocks = (E + TILE_EDGES - 1) / TILE_EDGES;
    gcn_spmm_scatter<<<blocks, BLOCK, 0, stream>>>(rows, cols, vals, embeds, out, E);
}